// VGG_Sym_29497835389127
// MI455X (gfx1250) — compile-verified
//
#include <hip/hip_runtime.h>
#include <hip/hip_bf16.h>

#define NC 512

typedef __attribute__((ext_vector_type(2))) float v2f;
typedef __attribute__((ext_vector_type(4))) float v4f;
typedef __attribute__((ext_vector_type(8))) float v8f;

// ---------------------------------------------------------------------------
// Stage 1: quantize x (8,3,67,67) f32 -> symbols (img,y,x,c) int32
// Binary search over sorted centroid_lut; tie goes to lower index (argmin).
// ---------------------------------------------------------------------------
__global__ void quantize_kernel(const float* __restrict__ x,
                                const float* __restrict__ cent,
                                int* __restrict__ s0)
{
    int idx = blockIdx.x * blockDim.x + threadIdx.x;
    const int total = 8 * 67 * 67 * 3;
    if (idx >= total) return;
    int c    = idx % 3;  int r = idx / 3;
    int xx   = r % 67;   r /= 67;
    int yy   = r % 67;   int img = r / 67;
    float v = x[((img * 3 + c) * 67 + yy) * 67 + xx];

    // lower_bound: first i with cent[i] >= v
    int lo = 0, hi = NC;
    while (lo < hi) {
        int mid = (lo + hi) >> 1;
        if (cent[mid] < v) lo = mid + 1; else hi = mid;
    }
    int i0 = (lo > 0)  ? lo - 1 : 0;
    int i1 = (lo < NC) ? lo     : NC - 1;
    float d0 = v - cent[i0]; d0 *= d0;
    float d1 = cent[i1] - v; d1 *= d1;
    s0[idx] = (d0 <= d1) ? i0 : i1;
}

// ---------------------------------------------------------------------------
// Stage 2: symbolic conv layer. One thread per (img, oy, ox, oc).
// Strictly-ordered fold: acc = conv_lut[sym0][w0];
//                        acc = add_lut[acc][conv_lut[symj][wj]] for j=1..
// Padding inserts symbol 0 (matches jnp.pad with zeros).
// Consecutive threads vary oc -> conv_lut row locality + coalesced W loads.
// Tables (2x1MB) live in the 192MB L2; this stage is gather-latency bound,
// so keep per-thread state tiny and occupancy high.
// ---------------------------------------------------------------------------
__global__ void symconv_kernel(const int* __restrict__ sin,
                               const int* __restrict__ W,
                               const int* __restrict__ conv_lut,
                               const int* __restrict__ add_lut,
                               const int* __restrict__ relu_lut,
                               int* __restrict__ sout,
                               int IH, int IW, int IC,
                               int OH, int OW, int OC,
                               int K, int ST, int PAD, int relu)
{
    int idx = blockIdx.x * blockDim.x + threadIdx.x;
    int total = 8 * OH * OW * OC;
    if (idx >= total) return;
    int oc = idx % OC; int r = idx / OC;
    int ox = r % OW;   r /= OW;
    int oy = r % OH;   int img = r / OH;

    const int* sbase = sin + img * IH * IW * IC;
    int acc = 0;
    int j = 0;
    for (int ky = 0; ky < K; ++ky) {
        int iy = oy * ST - PAD + ky;
        bool yok = (iy >= 0) && (iy < IH);
        for (int kx = 0; kx < K; ++kx) {
            int ix = ox * ST - PAD + kx;
            bool ok = yok && (ix >= 0) && (ix < IW);
            const int* srow = sbase + (iy * IW + ix) * IC;
            for (int c = 0; c < IC; ++c) {
                int sym = ok ? srow[c] : 0;
                int w   = W[j * OC + oc];
                int t   = conv_lut[(sym << 9) + w];
                acc     = (j == 0) ? t : add_lut[(acc << 9) + t];
                ++j;
            }
        }
    }
    if (relu) acc = relu_lut[acc];
    sout[idx] = acc;
}

// ---------------------------------------------------------------------------
// Stage 3: centroid decode + 2x2 mean pool -> padded 16x512 feature matrix
// (rows 8..15 zeroed so the WMMA M=16 tile is valid).
// ---------------------------------------------------------------------------
__global__ void pool_feats_kernel(const int* __restrict__ s,
                                  const float* __restrict__ cent,
                                  float* __restrict__ featsP)
{
    int idx = blockIdx.x * blockDim.x + threadIdx.x;
    if (idx >= 16 * 512) return;
    int c = idx % 512;
    int row = idx / 512;
    float v = 0.0f;
    if (row < 8) {
        int base = row * 2 * 2 * 512 + c;  // s layout (img, 2, 2, 512)
        v = 0.25f * (cent[s[base]] + cent[s[base + 512]] +
                     cent[s[base + 1024]] + cent[s[base + 1536]]);
    }
    featsP[idx] = v;
}

// ---------------------------------------------------------------------------
// Stage 4: FC via V_WMMA_F32_16X16X4_F32 (full fp32 precision).
// out(8,1000) = featsP(16x512) @ fc_w(1000x512)^T + fc_b, N tiled by 16.
// ISA 7.12.2 layouts:
//   A 16x4: lane<16 -> M=lane, vgpr0=K0, vgpr1=K1; lane>=16 -> K2,K3.
//   B 4x16: vgpr0 -> {K0 lanes0-15, K2 lanes16-31}, vgpr1 -> {K1, K3}.
//   C/D:    vgpr r -> M=r (lanes 0-15), M=r+8 (lanes 16-31); N = lane&15.
//
// Per 8-wide K block we issue one b128 load of A and one of B per lane and
// feed TWO WMMAs: call 0 consumes actual columns {8t+0,8t+1}|{8t+4,8t+5},
// call 1 consumes {8t+2,8t+3}|{8t+6,8t+7} (free relabeling of K columns —
// f32 K-accumulation order is not bitwise defined anyway). Lane float offset
// = 8t + (lane>=16 ? 4 : 0), always 16B aligned. No divergence in the loop:
// out-of-range N lanes clamp their fc_w row (results never stored).
// ---------------------------------------------------------------------------
__global__ __launch_bounds__(32)
void fc_wmma_kernel(const float* __restrict__ featsP,
                    const float* __restrict__ fc_w,
                    const float* __restrict__ fc_b,
                    float* __restrict__ out)
{
    int tile = blockIdx.x;            // N tile: columns tile*16 .. tile*16+15
    int lane = threadIdx.x;           // wave32, EXEC all 1s at WMMA issue
    int m    = lane & 15;
    int koff = (lane >> 4) * 4;       // 0 for lanes 0-15, 4 for lanes 16-31
    int n    = tile * 16 + m;
    int ncl  = (n < 1000) ? n : 999;  // clamp: junk lanes never stored

    const float* arow = featsP + m * 512 + koff;
    const float* brow = fc_w + ncl * 512 + koff;   // fc_w[n][k], row-major

    v8f acc = {};
    for (int k = 0; k < 512; k += 8) {
        v4f av = *(const v4f*)(arow + k);
        v4f bv = *(const v4f*)(brow + k);
        v2f a0; a0.x = av.x; a0.y = av.y;
        v2f b0; b0.x = bv.x; b0.y = bv.y;
        v2f a1; a1.x = av.z; a1.y = av.w;
        v2f b1; b1.x = bv.z; b1.y = bv.w;
        acc = __builtin_amdgcn_wmma_f32_16x16x4_f32(
            false, a0, false, b0, (short)0, acc, false, false);
        acc = __builtin_amdgcn_wmma_f32_16x16x4_f32(
            false, a1, false, b1, (short)0, acc, false, false);
    }

    // Lanes 0-15 hold M=0..7 (the real images) across acc[0..7].
    if (lane < 16 && n < 1000) {
        float bias = fc_b[n];
#pragma unroll
        for (int r = 0; r < 8; ++r)
            out[r * 1000 + n] = acc[r] + bias;
    }
}

// ---------------------------------------------------------------------------
// Launch
// ---------------------------------------------------------------------------
extern "C" void kernel_launch(void* const* d_in, const int* in_sizes, int n_in,
                              void* d_out, int out_size, void* d_ws, size_t ws_size,
                              hipStream_t stream)
{
    const float* x        = (const float*)d_in[0];
    const int*   w[17];
    for (int i = 0; i < 17; ++i) w[i] = (const int*)d_in[1 + i];
    const int*   conv_lut = (const int*)d_in[18];
    const int*   add_lut  = (const int*)d_in[19];
    const int*   relu_lut = (const int*)d_in[20];
    const float* cent     = (const float*)d_in[21];
    const float* fc_w     = (const float*)d_in[22];
    const float* fc_b     = (const float*)d_in[23];
    float*       out      = (float*)d_out;

    int*   bufA   = (int*)d_ws;
    int*   bufB   = (int*)((char*)d_ws + (1u << 20));
    float* featsP = (float*)((char*)d_ws + (2u << 20));

    // Stage 1: quantize into bufA as (img, y, x, c)
    {
        int total = 8 * 67 * 67 * 3;
        quantize_kernel<<<(total + 255) / 256, 256, 0, stream>>>(x, cent, bufA);
    }

    // Stage 2: the 17 symbolic conv layers (ping-pong bufA <-> bufB)
    struct Layer { int IH, IW, IC, OH, OW, OC, K, ST, PAD, relu; };
    const Layer L[17] = {
        {67, 67,   3, 16, 16,  64, 7, 4, 0, 1},  // 0
        {16, 16,  64, 16, 16,  64, 3, 1, 1, 1},  // 1
        {16, 16,  64, 16, 16,  64, 3, 1, 1, 0},  // 2
        {16, 16,  64, 16, 16,  64, 3, 1, 1, 1},  // 3
        {16, 16,  64, 16, 16,  64, 3, 1, 1, 0},  // 4
        {16, 16,  64,  8,  8, 128, 3, 2, 1, 1},  // 5
        { 8,  8, 128,  8,  8, 128, 3, 1, 1, 0},  // 6
        { 8,  8, 128,  8,  8, 128, 3, 1, 1, 1},  // 7
        { 8,  8, 128,  8,  8, 128, 3, 1, 1, 0},  // 8
        { 8,  8, 128,  4,  4, 256, 3, 2, 1, 1},  // 9
        { 4,  4, 256,  4,  4, 256, 3, 1, 1, 0},  // 10
        { 4,  4, 256,  4,  4, 256, 3, 1, 1, 1},  // 11
        { 4,  4, 256,  4,  4, 256, 3, 1, 1, 0},  // 12
        { 4,  4, 256,  2,  2, 512, 3, 2, 1, 1},  // 13
        { 2,  2, 512,  2,  2, 512, 3, 1, 1, 0},  // 14
        { 2,  2, 512,  2,  2, 512, 3, 1, 1, 1},  // 15
        { 2,  2, 512,  2,  2, 512, 3, 1, 1, 0},  // 16
    };

    int* cur = bufA;
    int* nxt = bufB;
    for (int i = 0; i < 17; ++i) {
        const Layer& l = L[i];
        int total = 8 * l.OH * l.OW * l.OC;
        symconv_kernel<<<(total + 255) / 256, 256, 0, stream>>>(
            cur, w[i], conv_lut, add_lut, relu_lut, nxt,
            l.IH, l.IW, l.IC, l.OH, l.OW, l.OC, l.K, l.ST, l.PAD, l.relu);
        int* t = cur; cur = nxt; nxt = t;
    }
    // Final symbols now in `cur` with shape (8, 2, 2, 512)

    // Stage 3: decode + mean-pool into padded 16x512 feature matrix
    pool_feats_kernel<<<(16 * 512 + 255) / 256, 256, 0, stream>>>(cur, cent, featsP);

    // Stage 4: FC via f32 WMMA. 63 tiles cover N=1008 >= 1000.
    fc_wmma_kernel<<<63, 32, 0, stream>>>(featsP, fc_w, fc_b, out);
}